// SkillGraphGNN_39608188403876
// MI455X (gfx1250) — compile-verified
//
#include <hip/hip_runtime.h>
#include <hip/hip_bf16.h>

typedef __attribute__((ext_vector_type(2))) float v2f;
typedef __attribute__((ext_vector_type(8))) float v8f;

#define TPB 256

// Guaranteed hardware fp32 scatter-add: non-returning GLOBAL_ATOMIC_ADD_F32
// (RT temporal, device scope via L2). Avoids any risk of a CAS-loop expansion.
__device__ __forceinline__ void atomic_add_f32(float* p, float v) {
#if defined(__gfx1250__)
  asm volatile("global_atomic_add_f32 %0, %1, off" : : "v"(p), "v"(v) : "memory");
#else
  atomicAdd(p, v);
#endif
}

// deg[i] = 1 (self loop), then += 1 per incoming edge
__global__ void gcn_deg_init(float* __restrict__ deg, int n) {
  int i = blockIdx.x * blockDim.x + threadIdx.x;
  if (i < n) deg[i] = 1.0f;
}

__global__ void gcn_deg_accum(const int* __restrict__ dst, float* __restrict__ deg, int E) {
  int e = blockIdx.x * blockDim.x + threadIdx.x;
  if (e < E) {
    int d = __builtin_nontemporal_load(&dst[e]);  // streamed once: keep out of L2
    atomic_add_f32(&deg[d], 1.0f);
  }
}

__global__ void gcn_dinv(const float* __restrict__ deg, float* __restrict__ dinv, int n) {
  int i = blockIdx.x * blockDim.x + threadIdx.x;
  if (i < n) dinv[i] = rsqrtf(deg[i]);  // deg >= 1 always (self-loops)
}

// C[M,N] = A[M,K] @ B[K,N], all row-major fp32.
// One wave32 per 16x16 C tile, V_WMMA_F32_16X16X4_F32 over K in steps of 4.
// Requires M%16==0, N%16==0, K%4==0 (holds: M=100000, N in {128,64}, K in {64,128}).
__global__ void gcn_wmma_gemm_f32(const float* __restrict__ A,
                                  const float* __restrict__ B,
                                  float* __restrict__ C,
                                  int M, int N, int K) {
  int wave = (int)((blockIdx.x * blockDim.x + threadIdx.x) >> 5);
  int lane = threadIdx.x & 31;
  int tilesN = N >> 4;
  int tilesM = M >> 4;
  if (wave >= tilesM * tilesN) return;  // wave-uniform: EXEC stays all-ones
  int tm = (wave / tilesN) << 4;
  int tn = (wave % tilesN) << 4;
  int half = lane >> 4;   // 0: lanes 0-15 (K+0,K+1), 1: lanes 16-31 (K+2,K+3)
  int l15  = lane & 15;

  // A 16x4 frag: lane holds A[tm+l15][k+2*half .. k+2*half+1] (contiguous -> b64 load)
  const float* arow = A + (size_t)(tm + l15) * K + 2 * half;
  // B 4x16 frag: lane holds B[k+2*half][tn+l15], B[k+2*half+1][tn+l15]
  const float* bcol = B + (size_t)(2 * half) * N + tn + l15;

  v8f acc = {};
  for (int k = 0; k < K; k += 4) {
    v2f a = *(const v2f*)(arow + k);
    v2f b;
    b.x = bcol[(size_t)k * N];
    b.y = bcol[(size_t)(k + 1) * N];
    acc = __builtin_amdgcn_wmma_f32_16x16x4_f32(false, a, false, b,
                                                (short)0, acc, false, false);
  }
  // D: VGPR r -> row (tm + r + 8*half), col (tn + l15)
  float* crow = C + (size_t)(tm + 8 * half) * N + tn + l15;
#pragma unroll
  for (int r = 0; r < 8; ++r) crow[(size_t)r * N] = acc[r];
}

// out[i,f] = h[i,f]*dinv[i]^2 (+ bias[f])  -- self-loop message, optional bias fold
__global__ void gcn_self_init(float* __restrict__ out, const float* __restrict__ h,
                              const float* __restrict__ dinv,
                              const float* __restrict__ bias, int n, int logF) {
  int i = blockIdx.x * blockDim.x + threadIdx.x;
  int total = n << logF;
  if (i >= total) return;
  int node = i >> logF;
  int f = i & ((1 << logF) - 1);
  float di = dinv[node];
  float v = h[i] * di * di;
  if (bias) v += bias[f];
  out[i] = v;
}

__global__ void gcn_bias_relu(float* __restrict__ buf, const float* __restrict__ bias,
                              int n, int logF) {
  int i = blockIdx.x * blockDim.x + threadIdx.x;
  if (i >= (n << logF)) return;
  buf[i] = fmaxf(buf[i] + bias[i & ((1 << logF) - 1)], 0.0f);
}

// One wave32 per edge; each lane moves VEC consecutive floats (F = 32*VEC).
// 128-bit/64-bit gather loads (RT: h stays L2-resident) + hardware f32 atomics.
// Edge-index loads are NT so the 26MB edge stream doesn't evict h/agg from L2.
template <int VEC>
__global__ void gcn_edge_scatter(const int* __restrict__ src, const int* __restrict__ dst,
                                 const float* __restrict__ dinv,
                                 const float* __restrict__ h,
                                 float* __restrict__ out, int E) {
  constexpr int F = 32 * VEC;
  int wid = (int)((blockIdx.x * blockDim.x + threadIdx.x) >> 5);
  int lane = threadIdx.x & 31;
  if (wid >= E) return;  // wave-uniform
  int s = __builtin_nontemporal_load(&src[wid]);
  int d = __builtin_nontemporal_load(&dst[wid]);
  float norm = dinv[s] * dinv[d];
  const float* hp = h + (size_t)s * F + lane * VEC;
  float* op = out + (size_t)d * F + lane * VEC;
  if constexpr (VEC == 4) {
    float4 v = *(const float4*)hp;  // global_load_b128
    atomic_add_f32(op + 0, v.x * norm);
    atomic_add_f32(op + 1, v.y * norm);
    atomic_add_f32(op + 2, v.z * norm);
    atomic_add_f32(op + 3, v.w * norm);
  } else {
    float2 v = *(const float2*)hp;  // global_load_b64
    atomic_add_f32(op + 0, v.x * norm);
    atomic_add_f32(op + 1, v.y * norm);
  }
}

static inline int cdiv_l(long a, long b) { return (int)((a + b - 1) / b); }

extern "C" void kernel_launch(void* const* d_in, const int* in_sizes, int n_in,
                              void* d_out, int out_size, void* d_ws, size_t ws_size,
                              hipStream_t stream) {
  (void)n_in; (void)out_size; (void)ws_size;
  const float* x  = (const float*)d_in[0];
  const int*   ei = (const int*)d_in[1];
  const float* W1 = (const float*)d_in[2];
  const float* b1 = (const float*)d_in[3];
  const float* W2 = (const float*)d_in[4];
  const float* b2 = (const float*)d_in[5];
  float* out = (float*)d_out;

  const int Nn = in_sizes[0] / 64;   // 100000 nodes (x is [N,64])
  const int E  = in_sizes[1] / 2;    // 3.2M edges
  const int* src = ei;               // edge_index row 0
  const int* dst = ei + E;           // edge_index row 1

  // workspace layout (floats): deg[N] | dinv[N] | h[N*128] | agg[N*128]
  float* deg  = (float*)d_ws;
  float* dinv = deg + Nn;
  float* h    = dinv + Nn;                 // layer1 pre-agg; reused as layer2 GEMM out (N*64)
  float* agg  = h + (size_t)Nn * 128;      // layer1 accumulator / layer2 GEMM input

  dim3 blk(TPB);

  // degrees + normalization
  gcn_deg_init<<<cdiv_l(Nn, TPB), blk, 0, stream>>>(deg, Nn);
  gcn_deg_accum<<<cdiv_l(E, TPB), blk, 0, stream>>>(dst, deg, E);
  gcn_dinv<<<cdiv_l(Nn, TPB), blk, 0, stream>>>(deg, dinv, Nn);

  // ---- layer 1: h = x @ W1  (M=N, K=64, N=128) ----
  {
    long waves = (long)(Nn / 16) * (128 / 16);
    gcn_wmma_gemm_f32<<<cdiv_l(waves * 32, TPB), blk, 0, stream>>>(x, W1, h, Nn, 128, 64);
  }
  gcn_self_init<<<cdiv_l((long)Nn * 128, TPB), blk, 0, stream>>>(agg, h, dinv, nullptr, Nn, 7);
  gcn_edge_scatter<4><<<cdiv_l((long)E * 32, TPB), blk, 0, stream>>>(src, dst, dinv, h, agg, E);
  gcn_bias_relu<<<cdiv_l((long)Nn * 128, TPB), blk, 0, stream>>>(agg, b1, Nn, 7);

  // ---- layer 2: h2 = relu_h @ W2  (M=N, K=128, N=64) -> reuse h buffer ----
  {
    long waves = (long)(Nn / 16) * (64 / 16);
    gcn_wmma_gemm_f32<<<cdiv_l(waves * 32, TPB), blk, 0, stream>>>(agg, W2, h, Nn, 64, 128);
  }
  gcn_self_init<<<cdiv_l((long)Nn * 64, TPB), blk, 0, stream>>>(out, h, dinv, b2, Nn, 6);
  gcn_edge_scatter<2><<<cdiv_l((long)E * 32, TPB), blk, 0, stream>>>(src, dst, dinv, h, out, E);
}